// GraphConvEmb_53867479826619
// MI455X (gfx1250) — compile-verified
//
#include <hip/hip_runtime.h>
#include <hip/hip_bf16.h>
#include <math.h>

#define N_LAYERS 5
#define HID 64
#define N_FF 512
#define N_NODES 50000
#define N_EDGES 800000

typedef float v2f __attribute__((ext_vector_type(2)));
typedef float v8f __attribute__((ext_vector_type(8)));

// ---------------- embedding gather: h[n] = vert_emb[x[n]] ----------------
__global__ void k_embed(const int* __restrict__ x, const float* __restrict__ vemb,
                        float* __restrict__ h) {
    int idx = blockIdx.x * blockDim.x + threadIdx.x;     // over N_NODES * (HID/4)
    if (idx >= N_NODES * (HID / 4)) return;
    int n  = idx / (HID / 4);
    int c4 = idx % (HID / 4);
    const float4* s = (const float4*)(vemb + (size_t)x[n] * HID);
    float4* d = (float4*)(h + (size_t)n * HID);
    d[c4] = s[c4];
}

// ---------------- Ml = edge_emb_w (4x64) @ edge_lin_w[l] (64x64) ----------------
__global__ void k_ml(const float* __restrict__ ee, const float* __restrict__ elin,
                     float* __restrict__ Ml) {
    int t = threadIdx.x;            // 256 threads: (j, c)
    int j = t >> 6, c = t & 63;
    float s = 0.f;
#pragma unroll
    for (int k = 0; k < HID; ++k) s += ee[j * HID + k] * elin[k * HID + c];
    Ml[t] = s;
}

// ---------------- zero fill ----------------
__global__ void k_zero(float* __restrict__ p, int n) {
    int i = blockIdx.x * blockDim.x + threadIdx.x;
    if (i < n) p[i] = 0.f;
}

// ------- edge kernel: agg[dst] += h[src] * (edge_attr[e] @ Ml) (8 cols/thread) -------
__global__ void __launch_bounds__(256) k_edge(const int* __restrict__ src,
                                              const int* __restrict__ dst,
                                              const float* __restrict__ eattr,
                                              const float* __restrict__ Ml,
                                              const float* __restrict__ h,
                                              float* __restrict__ agg) {
    __shared__ float sMl[4 * HID];
    sMl[threadIdx.x] = Ml[threadIdx.x];                 // blockDim == 256 == 4*HID
    __syncthreads();
    int gid = blockIdx.x * blockDim.x + threadIdx.x;
    int e = gid >> 3;
    if (e >= N_EDGES) return;
    int part = gid & 7;
    int c0 = part * 8;
    float4 a = *(const float4*)(eattr + (size_t)e * 4);
    int s = src[e], d = dst[e];
    const float* hs = h + (size_t)s * HID;
    float* ad = agg + (size_t)d * HID;
#pragma unroll
    for (int j = 0; j < 8; ++j) {
        int c = c0 + j;
        float v = a.x * sMl[c] + a.y * sMl[HID + c] + a.z * sMl[2 * HID + c]
                + a.w * sMl[3 * HID + c];
        unsafeAtomicAdd(&ad[c], hs[c] * v);
    }
}

// ---- node kernel: hnext = relu(agg@relw + relb + hcur@rootw) + hcur@resw (WMMA) ----
__global__ void __launch_bounds__(256) k_node(const float* __restrict__ hcur,
                                              const float* __restrict__ agg,
                                              const float* __restrict__ relw,
                                              const float* __restrict__ relb,
                                              const float* __restrict__ rootw,
                                              const float* __restrict__ resw,
                                              float* __restrict__ hnext) {
    const int wave = threadIdx.x >> 5;
    const int lane = threadIdx.x & 31;
    const int mtile = blockIdx.x * 2 + (wave >> 2);     // 2 m-tiles per block
    const int ntile = wave & 3;                          // 64/16 = 4 n-tiles
    if (mtile >= N_NODES / 16) return;                   // uniform per wave
    const int half = lane >> 4;
    const int l16 = lane & 15;
    const int arow = mtile * 16 + l16;                   // A row for this lane
    const int col = ntile * 16 + l16;                    // B/D column for this lane
    v8f accRel = {0.f, 0.f, 0.f, 0.f, 0.f, 0.f, 0.f, 0.f};
    v8f accRoot = accRel, accRes = accRel;
    for (int k = 0; k < HID; k += 4) {
        const int ka = k + 2 * half;                     // K pair for this half-wave
        v2f aAgg, aH, bRel, bRoot, bRes;
        aAgg.x = agg[(size_t)arow * HID + ka];
        aAgg.y = agg[(size_t)arow * HID + ka + 1];
        aH.x = hcur[(size_t)arow * HID + ka];
        aH.y = hcur[(size_t)arow * HID + ka + 1];
        bRel.x = relw[(size_t)ka * HID + col];
        bRel.y = relw[(size_t)(ka + 1) * HID + col];
        bRoot.x = rootw[(size_t)ka * HID + col];
        bRoot.y = rootw[(size_t)(ka + 1) * HID + col];
        bRes.x = resw[(size_t)ka * HID + col];
        bRes.y = resw[(size_t)(ka + 1) * HID + col];
        accRel = __builtin_amdgcn_wmma_f32_16x16x4_f32(false, aAgg, false, bRel,
                                                       (short)0, accRel, false, false);
        accRoot = __builtin_amdgcn_wmma_f32_16x16x4_f32(false, aH, false, bRoot,
                                                        (short)0, accRoot, false, false);
        accRes = __builtin_amdgcn_wmma_f32_16x16x4_f32(false, aH, false, bRes,
                                                       (short)0, accRes, false, false);
    }
    const float bias = relb[col];
#pragma unroll
    for (int r = 0; r < 8; ++r) {
        const int orow = mtile * 16 + r + 8 * half;      // C/D layout row
        float o = accRel[r] + accRoot[r] + bias;
        o = fmaxf(o, 0.f);
        hnext[(size_t)orow * HID + col] = o + accRes[r];
    }
}

// ---------------- head: out[n] = sum_f gelu(h@W1 + b1)[n,f] * w2[f] + b2 ----------------
__global__ void k_head_init(const float* __restrict__ b2, float* __restrict__ out) {
    int i = blockIdx.x * blockDim.x + threadIdx.x;
    if (i < N_NODES) out[i] = b2[0];
}

__global__ void __launch_bounds__(256) k_head(const float* __restrict__ h,
                                              const float* __restrict__ w1,
                                              const float* __restrict__ b1,
                                              const float* __restrict__ w2,
                                              float* __restrict__ out) {
    const int wave = threadIdx.x >> 5;
    const int lane = threadIdx.x & 31;
    const int id = blockIdx.x * 8 + wave;                // 3125 * 32 waves exactly
    const int ftile = id & 31;                           // 512/16 = 32 ff tiles
    const int mtile = id >> 5;
    const int half = lane >> 4;
    const int l16 = lane & 15;
    const int arow = mtile * 16 + l16;
    const int col = ftile * 16 + l16;
    v8f acc = {0.f, 0.f, 0.f, 0.f, 0.f, 0.f, 0.f, 0.f};
    for (int k = 0; k < HID; k += 4) {
        const int ka = k + 2 * half;
        v2f a, b;
        a.x = h[(size_t)arow * HID + ka];
        a.y = h[(size_t)arow * HID + ka + 1];
        b.x = w1[(size_t)ka * N_FF + col];
        b.y = w1[(size_t)(ka + 1) * N_FF + col];
        acc = __builtin_amdgcn_wmma_f32_16x16x4_f32(false, a, false, b,
                                                    (short)0, acc, false, false);
    }
    const float bb = b1[col];
    const float ww = w2[col];
#pragma unroll
    for (int r = 0; r < 8; ++r) {
        float v = acc[r] + bb;
        float g = 0.5f * v * (1.f + erff(v * 0.70710678118654752f));   // exact erf GELU
        float p = g * ww;
        p += __shfl_xor(p, 1, 32);
        p += __shfl_xor(p, 2, 32);
        p += __shfl_xor(p, 4, 32);
        p += __shfl_xor(p, 8, 32);                       // reduce within 16-lane group
        if (l16 == 0) unsafeAtomicAdd(&out[mtile * 16 + r + 8 * half], p);
    }
}

extern "C" void kernel_launch(void* const* d_in, const int* in_sizes, int n_in,
                              void* d_out, int out_size, void* d_ws, size_t ws_size,
                              hipStream_t stream) {
    const int* x = (const int*)d_in[0];
    const int* ei = (const int*)d_in[1];
    const int* src = ei;
    const int* dst = ei + N_EDGES;
    const float* eattr = (const float*)d_in[2];
    const float* vemb = (const float*)d_in[3];
    const float* eew = (const float*)d_in[4];
    const float* elin = (const float*)d_in[5];
    const float* relw = (const float*)d_in[6];
    const float* relb = (const float*)d_in[7];
    const float* rootw = (const float*)d_in[8];
    const float* resw = (const float*)d_in[9];
    const float* w1 = (const float*)d_in[10];
    const float* b1 = (const float*)d_in[11];
    const float* w2 = (const float*)d_in[12];
    const float* b2 = (const float*)d_in[13];
    float* out = (float*)d_out;

    const size_t NH = (size_t)N_NODES * HID;             // 3.2M floats
    float* ws = (float*)d_ws;
    float* hA = ws;
    float* hB = ws + NH;
    float* agg = ws + 2 * NH;
    float* Ml = ws + 3 * NH;                             // 256 floats

    // 1) embedding
    {
        int total = N_NODES * (HID / 4);
        k_embed<<<(total + 255) / 256, 256, 0, stream>>>(x, vemb, hA);
    }

    // 2) layers
    float* cur = hA;
    float* nxt = hB;
    for (int l = 0; l < N_LAYERS; ++l) {
        k_ml<<<1, 256, 0, stream>>>(eew, elin + (size_t)l * HID * HID, Ml);
        k_zero<<<(int)((NH + 255) / 256), 256, 0, stream>>>(agg, (int)NH);
        {
            int total = N_EDGES * 8;
            k_edge<<<(total + 255) / 256, 256, 0, stream>>>(src, dst, eattr, Ml, cur, agg);
        }
        {
            int mtiles = N_NODES / 16;                   // 3125
            int blocks = (mtiles + 1) / 2;               // 2 m-tiles (8 waves) per block
            k_node<<<blocks, 256, 0, stream>>>(cur, agg,
                                               relw + (size_t)l * HID * HID,
                                               relb + (size_t)l * HID,
                                               rootw + (size_t)l * HID * HID,
                                               resw + (size_t)l * HID * HID,
                                               nxt);
        }
        float* t = cur; cur = nxt; nxt = t;
    }

    // 3) head
    k_head_init<<<(N_NODES + 255) / 256, 256, 0, stream>>>(b2, out);
    {
        int waves = (N_NODES / 16) * (N_FF / 16);        // 100000
        int blocks = waves / 8;                          // 12500 exactly
        k_head<<<blocks, 256, 0, stream>>>(cur, w1, b1, w2, out);
    }
}